// DressedQuantumCircuit_24902220383012
// MI455X (gfx1250) — compile-verified
//
#include <hip/hip_runtime.h>
#include <math.h>

typedef __attribute__((ext_vector_type(2))) float v2f;
typedef __attribute__((ext_vector_type(8))) float v8f;

#define BLOCK 256

// M[m][k] = prod_j U_j[m_j][k_j], U = [[c,-s],[s,c]], bit j=0 is MSB of 4-bit index
__device__ __forceinline__ float mat_entry(int m, int k, const float cw[4], const float sw[4]) {
    float p = 1.0f;
#pragma unroll
    for (int j = 0; j < 4; ++j) {
        int mb = (m >> (3 - j)) & 1;
        int kb = (k >> (3 - j)) & 1;
        float u = (mb == kb) ? cw[j] : (mb ? sw[j] : -sw[j]);
        p *= u;
    }
    return p;
}

__global__ __launch_bounds__(BLOCK)
void dqc_wmma_kernel(const float* __restrict__ feat,
                     const float* __restrict__ qp,
                     float* __restrict__ out) {
    constexpr float PI_4 = 0.78539816339744830962f;
    // cos/sin of k*pi/4 for k in [-2..3], index k+2
    constexpr float CRZ_C[6] = {0.0f, 0.70710678118654752f, 1.0f,
                                0.70710678118654752f, 0.0f, -0.70710678118654752f};
    constexpr float CRZ_S[6] = {-1.0f, -0.70710678118654752f, 0.0f,
                                0.70710678118654752f, 1.0f, 0.70710678118654752f};

    __shared__ float s_re[BLOCK][17];   // stride 17 -> conflict-free (17 coprime 64)
    __shared__ float s_im[BLOCK][17];

    const int tid  = threadIdx.x;
    const int lane = tid & 31;
    const int wv   = tid >> 5;
    const int r    = lane & 15;     // row within 16 / element column within tile
    const int hi   = lane >> 4;     // lower/upper half-wave
    const int elem = blockIdx.x * BLOCK + tid;

    // ---- per-element angles: theta_q/2 = tanh(x)*pi/4 ----
    float4 f = reinterpret_cast<const float4*>(feat)[elem];
    float th[4] = {f.x, f.y, f.z, f.w};
    float cc[4], ss[4];
#pragma unroll
    for (int q = 0; q < 4; ++q) {
        float a = tanhf(th[q]) * PI_4;
        __sincosf(a, &ss[q], &cc[q]);
    }

    // ---- diagonal part of circuit (H + RZ + CRZ), bit-permuted to s' = (q1 q3 q0 q2) ----
#pragma unroll
    for (int sp = 0; sp < 16; ++sp) {
        const int q1 = (sp >> 3) & 1;
        const int q3 = (sp >> 2) & 1;
        const int q0 = (sp >> 1) & 1;
        const int q2 = sp & 1;
        const int kk = (q0 ? (q1 ? 1 : -1) : 0)
                     + (q1 ? (q2 ? 1 : -1) : 0)
                     + (q2 ? (q3 ? 1 : -1) : 0);     // CRZ phase in units of pi/4
        float re = 0.25f * CRZ_C[kk + 2];
        float im = 0.25f * CRZ_S[kk + 2];
        const int bits[4] = {q0, q1, q2, q3};
#pragma unroll
        for (int q = 0; q < 4; ++q) {                 // multiply by e^{+/- i theta_q/2}
            float br = cc[q];
            float bi = bits[q] ? ss[q] : -ss[q];
            float nr = re * br - im * bi;
            float ni = re * bi + im * br;
            re = nr; im = ni;
        }
        s_re[tid][sp] = re;
        s_im[tid][sp] = im;
    }

    // ---- batch-shared RY layer as WMMA A-matrix (16x4 f32 layout, 4 K-chunks) ----
    float cw[4], sw[4];
#pragma unroll
    for (int j = 0; j < 4; ++j) __sincosf(qp[j] * 0.5f, &sw[j], &cw[j]);
    v2f A[4];
#pragma unroll
    for (int c = 0; c < 4; ++c) {
        A[c].x = mat_entry(r, 4 * c + 2 * hi,     cw, sw);  // VGPR0: K = 4c + 2*hi
        A[c].y = mat_entry(r, 4 * c + 2 * hi + 1, cw, sw);  // VGPR1: K = 4c + 2*hi + 1
    }

    __syncthreads();

    // ---- two 16-element tiles per wave: D(16x16) = M(16x16) @ S(16x16), re & im ----
#pragma unroll
    for (int t = 0; t < 2; ++t) {
        const int n = wv * 32 + t * 16 + r;   // element-in-block this lane reads (B column)
        v8f accR = {0.f, 0.f, 0.f, 0.f, 0.f, 0.f, 0.f, 0.f};
        v8f accI = {0.f, 0.f, 0.f, 0.f, 0.f, 0.f, 0.f, 0.f};
#pragma unroll
        for (int c = 0; c < 4; ++c) {
            const int k0 = 4 * c + 2 * hi;    // B 4x16 layout mirrors A
            v2f bR; bR.x = s_re[n][k0]; bR.y = s_re[n][k0 + 1];
            v2f bI; bI.x = s_im[n][k0]; bI.y = s_im[n][k0 + 1];
            accR = __builtin_amdgcn_wmma_f32_16x16x4_f32(false, A[c], false, bR,
                                                         (short)0, accR, false, false);
            accI = __builtin_amdgcn_wmma_f32_16x16x4_f32(false, A[c], false, bI,
                                                         (short)0, accI, false, false);
        }

        // probs + signed Z-expectation partials over this lane's 8 states (m = v + 8*hi)
        float e0 = 0.f, e1 = 0.f, e2 = 0.f, e3 = 0.f;
#pragma unroll
        for (int v = 0; v < 8; ++v) {
            float p = accR[v] * accR[v] + accI[v] * accI[v];
            int m = v + 8 * hi;
            e0 += ((m >> 3) & 1) ? -p : p;
            e1 += ((m >> 2) & 1) ? -p : p;
            e2 += ((m >> 1) & 1) ? -p : p;
            e3 += ( m       & 1) ? -p : p;
        }
        // combine the other half of the states from the partner half-wave
        e0 += __shfl_xor(e0, 16, 32);
        e1 += __shfl_xor(e1, 16, 32);
        e2 += __shfl_xor(e2, 16, 32);
        e3 += __shfl_xor(e3, 16, 32);

        if (hi == t) {   // lanes 0-15 store tile 0, lanes 16-31 store tile 1
            int oe = blockIdx.x * BLOCK + wv * 32 + t * 16 + r;
            reinterpret_cast<float4*>(out)[oe] = make_float4(e0, e1, e2, e3);
        }
    }
}

extern "C" void kernel_launch(void* const* d_in, const int* in_sizes, int n_in,
                              void* d_out, int out_size, void* d_ws, size_t ws_size,
                              hipStream_t stream) {
    const float* feat = (const float*)d_in[0];   // (BATCH, 4) float32
    const float* qp   = (const float*)d_in[1];   // (4,) float32
    float* out        = (float*)d_out;           // (BATCH, 4) float32
    const int batch   = in_sizes[0] / 4;         // 524288
    dqc_wmma_kernel<<<batch / BLOCK, BLOCK, 0, stream>>>(feat, qp, out);
}